// UniversalBiphasicAxonMapModule_79121887527078
// MI455X (gfx1250) — compile-verified
//
#include <hip/hip_runtime.h>
#include <math.h>

typedef __attribute__((ext_vector_type(2))) float v2f;
typedef __attribute__((ext_vector_type(8))) float v8f;

#define LOG2E 1.44269504088896340736f
#define NB 2
#define NP 10000
#define NL 64
#define NE 100
#define WAVES_PER_BLOCK 8

// One wave per pixel p. Each wave computes out[b][p] for BOTH batches from a
// single streaming pass over d2_el[p, :, :] (256 MB read exactly once).
// Sum over electrodes is done on the matrix pipe: V_WMMA_F32_16X16X4_F32 with
// B == ones, so every column of D holds the f32 row-sum (no precision loss).
__global__ __launch_bounds__(256)
void axon_map_kernel(const float* __restrict__ stim,    // (B,E,3) freq,amp,pdur
                     const float* __restrict__ params,  // (B,12)  rho,axlam,a0..a9
                     const float* __restrict__ axon,    // (P,L,3) ax,ay,sens
                     const float* __restrict__ d2_el,   // (P,L,E)
                     float* __restrict__ out)           // (B,P)
{
    // ---- per-(b,e) coefficients in LDS: (negc1, c2s, Fb, 0), log2e folded in
    __shared__ float4 coef[NB][NE];
    const int tid = threadIdx.x;
    if (tid < NB * NE) {
        const int b = tid / NE, e = tid % NE;
        const float freq = stim[(b * NE + e) * 3 + 0];
        const float amp  = stim[(b * NE + e) * 3 + 1];
        const float pdur = stim[(b * NE + e) * 3 + 2];
        const float* pp = params + b * 12;
        const float rho = pp[0], axlam = pp[1];
        const float a0 = pp[2], a1 = pp[3], a2 = pp[4], a3 = pp[5];
        const float a5 = pp[7], a6 = pp[8], a7 = pp[9], a8 = pp[10], a9 = pp[11];

        const float scaled = (a1 + a0 * pdur) * amp;
        float Fb = a2 * scaled + a3 * freq;
        if (!(scaled > 0.25f)) Fb = 0.0f;   // AMP_CUTOFF
        if (!(amp > 0.0f))     Fb = 0.0f;   // !LIKE_JAX
        const float Fsize = fmaxf(a5 * scaled + a6, 100.0f / (rho * rho));
        const float pw = __builtin_amdgcn_exp2f(a8 * __builtin_amdgcn_logf(pdur)); // pdur^a8
        const float Fstreak = fmaxf(a9 - a7 * pw, 100.0f / (axlam * axlam));
        const float negc1 = -LOG2E / (2.0f * rho * rho * Fsize);
        const float c2s   =  LOG2E / (axlam * axlam * Fstreak);
        coef[b][e] = make_float4(negc1, c2s, Fb, 0.0f);
    }
    __syncthreads();

    const int lane = tid & 31;
    const int hf   = lane >> 4;   // half-wave: selects e-pair {0,1} vs {2,3}
    const int m    = lane & 15;   // row within 16-row tile
    const int p    = blockIdx.x * WAVES_PER_BLOCK + (tid >> 5);

    // 4 l-tiles of 16 rows each cover all L=64 segments of pixel p.
    float sensv[4];
    long  rowoff[4];
    #pragma unroll
    for (int t = 0; t < 4; ++t) {
        const long row = (long)p * NL + t * 16 + m;
        sensv[t]  = axon[row * 3 + 2];
        rowoff[t] = row * NE + hf * 2;     // float2 per lane per e-block
    }

    const v2f ONES = {1.0f, 1.0f};
    v8f acc[NB][4];
    #pragma unroll
    for (int b = 0; b < NB; ++b)
        #pragma unroll
        for (int t = 0; t < 4; ++t)
            acc[b][t] = (v8f){0.f, 0.f, 0.f, 0.f, 0.f, 0.f, 0.f, 0.f};

    for (int eb = 0; eb < NE / 4; ++eb) {
        const int e0 = eb * 4 + hf * 2;
        const float4 c00 = coef[0][e0], c01 = coef[0][e0 + 1];   // batch 0
        const float4 c10 = coef[1][e0], c11 = coef[1][e0 + 1];   // batch 1
        #pragma unroll
        for (int t = 0; t < 4; ++t) {
            const v2f d2 = *(const v2f*)(d2_el + rowoff[t] + eb * 4);
            const float s = sensv[t];
            // batch 0: w = Fb * 2^(d2*negc1 + sens*c2s)
            v2f A0;
            A0.x = c00.z * __builtin_amdgcn_exp2f(__builtin_fmaf(d2.x, c00.x, s * c00.y));
            A0.y = c01.z * __builtin_amdgcn_exp2f(__builtin_fmaf(d2.y, c01.x, s * c01.y));
            acc[0][t] = __builtin_amdgcn_wmma_f32_16x16x4_f32(
                false, A0, false, ONES, (short)0, acc[0][t], false, false);
            // batch 1 (reuses the same d2 load: halves HBM traffic)
            v2f A1;
            A1.x = c10.z * __builtin_amdgcn_exp2f(__builtin_fmaf(d2.x, c10.x, s * c10.y));
            A1.y = c11.z * __builtin_amdgcn_exp2f(__builtin_fmaf(d2.y, c11.x, s * c11.y));
            acc[1][t] = __builtin_amdgcn_wmma_f32_16x16x4_f32(
                false, A1, false, ONES, (short)0, acc[1][t], false, false);
        }
    }

    // acc[b][t][i]: lanes 0-15 hold row M=i (l = t*16+i), lanes 16-31 hold M=8+i.
    // All D columns are identical (B == ones) -> values uniform per half-wave.
    #pragma unroll
    for (int b = 0; b < NB; ++b) {
        float pm = 0.0f;  // sums are >= 0
        #pragma unroll
        for (int t = 0; t < 4; ++t) {
            const v8f c = acc[b][t];
            float mx = c[0];
            #pragma unroll
            for (int i = 1; i < 8; ++i) mx = fmaxf(mx, c[i]);
            mx = fmaxf(mx, __shfl_xor(mx, 16, 32));  // merge the two row-halves
            pm = fmaxf(pm, mx);
        }
        pm = (pm > 0.001f) ? pm : 0.0f;              // THRESH_PERCEPT
        if (lane == 0) out[b * NP + p] = pm;
    }
}

extern "C" void kernel_launch(void* const* d_in, const int* in_sizes, int n_in,
                              void* d_out, int out_size, void* d_ws, size_t ws_size,
                              hipStream_t stream) {
    (void)in_sizes; (void)n_in; (void)d_ws; (void)ws_size; (void)out_size;
    const float* stim    = (const float*)d_in[0];   // (2,100,3)
    const float* params  = (const float*)d_in[1];   // (2,12)
    const float* axon    = (const float*)d_in[2];   // (10000,64,3)
    const float* d2_el   = (const float*)d_in[3];   // (10000,64,100)
    float* out = (float*)d_out;                     // (2,100,100) flat

    const int blocks = NP / WAVES_PER_BLOCK;        // 1250 blocks * 8 waves = 10000 pixels
    hipLaunchKernelGGL(axon_map_kernel, dim3(blocks), dim3(256), 0, stream,
                       stim, params, axon, d2_el, out);
}